// LiftingDWT1D_12378095747844
// MI455X (gfx1250) — compile-verified
//
#include <hip/hip_runtime.h>

// Problem constants from the reference
#define BDIM 16
#define CDIM 64
#define HDIM 256
#define WDIM 256
#define WHALF 128

constexpr int ROWS_PER_BLOCK  = 4;   // rows of W=256 handled per block
constexpr int THREADS_PER_ROW = 64;  // each thread -> 2 output positions
constexpr int BLOCK           = ROWS_PER_BLOCK * THREADS_PER_ROW; // 256 (8 waves)
constexpr long long TOTAL_ROWS = (long long)BDIM * CDIM * HDIM;   // 262144

#if defined(__gfx1250__) && __has_builtin(__builtin_amdgcn_global_load_async_to_lds_b128)
#define USE_ASYNC_LDS 1
#else
#define USE_ASYNC_LDS 0
#endif

#if USE_ASYNC_LDS
typedef int v4i_vec __attribute__((vector_size(16)));
typedef __attribute__((address_space(1))) v4i_vec* global_v4i_ptr;
typedef __attribute__((address_space(3))) v4i_vec* lds_v4i_ptr;
#endif

__global__ __launch_bounds__(BLOCK)
void lifting_dwt1d_kernel(const float* __restrict__ x,
                          const float* __restrict__ Pw, const float* __restrict__ Pb,
                          const float* __restrict__ Uw, const float* __restrict__ Ub,
                          float* __restrict__ out_s, float* __restrict__ out_d)
{
    __shared__ float xrow[ROWS_PER_BLOCK][WDIM];   // staged input rows (4 KB)
    __shared__ float drow[ROWS_PER_BLOCK][WHALF];  // d for the s-stencil  (2 KB)

    const int tid = threadIdx.x;
    const int r   = tid >> 6;   // row within block
    const int t   = tid & 63;   // thread within row; owns w0=2t, w1=2t+1

    const long long row = (long long)blockIdx.x * ROWS_PER_BLOCK + r; // < TOTAL_ROWS
    const int c = (int)((row >> 8) & (CDIM - 1));  // (row / H) % C, H=256

    const float* gsrc = x + row * WDIM + 4 * t;    // 16B chunk of this row
    float* ldst = &xrow[r][4 * t];

    // ---- Stage the full 256-float row into LDS (gfx1250 async path) ----
#if USE_ASYNC_LDS
    __builtin_amdgcn_global_load_async_to_lds_b128(
        (global_v4i_ptr)gsrc,
        (lds_v4i_ptr)ldst,
        /*imm offset*/0, /*cpol*/0);
  #if __has_builtin(__builtin_amdgcn_s_wait_asynccnt)
    __builtin_amdgcn_s_wait_asynccnt(0);
  #else
    asm volatile("s_wait_asynccnt 0" ::: "memory");
  #endif
#else
    *(float4*)ldst = *(const float4*)gsrc;
#endif
    __syncthreads();

    // This thread's interleaved quad: e[2t], o[2t], e[2t+1], o[2t+1]
    const float4 q = *(const float4*)(&xrow[r][4 * t]);
    const float e0 = q.x, o0 = q.y, e1 = q.z, o1 = q.w;
    // Wrap neighbors: e[j] lives at xrow[2j]
    const float em1 = xrow[r][(4 * t - 2) & 255];  // e[(2t-1) & 127]
    const float ep2 = xrow[r][(4 * t + 4) & 255];  // e[(2t+2) & 127]

    const float P0 = Pw[3 * c], P1 = Pw[3 * c + 1], P2 = Pw[3 * c + 2], pb = Pb[c];

    // d = o - (P (*) e + pb), circular 3-tap cross-correlation
    const float d0 = o0 - (fmaf(P0, em1, fmaf(P1, e0, fmaf(P2, e1,  pb))));
    const float d1 = o1 - (fmaf(P0, e0,  fmaf(P1, e1, fmaf(P2, ep2, pb))));

    const int w0 = 2 * t;
    *(float2*)(&drow[r][w0]) = make_float2(d0, d1);
    __syncthreads();

    const float dm1 = drow[r][(w0 - 1) & 127];
    const float dp1 = drow[r][(w0 + 2) & 127];

    const float U0 = Uw[3 * c], U1 = Uw[3 * c + 1], U2 = Uw[3 * c + 2], ub = Ub[c];

    // s = e + (U (*) d + ub)
    const float s0 = e0 + fmaf(U0, dm1, fmaf(U1, d0, fmaf(U2, d1,  ub)));
    const float s1 = e1 + fmaf(U0, d0,  fmaf(U1, d1, fmaf(U2, dp1, ub)));

    const long long obase = row * WHALF + w0;
    *(float2*)(out_s + obase) = make_float2(s0, s1);
    *(float2*)(out_d + obase) = make_float2(d0, d1);
}

extern "C" void kernel_launch(void* const* d_in, const int* in_sizes, int n_in,
                              void* d_out, int out_size, void* d_ws, size_t ws_size,
                              hipStream_t stream) {
    const float* x  = (const float*)d_in[0];
    const float* Pw = (const float*)d_in[1];
    const float* Pb = (const float*)d_in[2];
    const float* Uw = (const float*)d_in[3];
    const float* Ub = (const float*)d_in[4];
    // d_in[5] is `dim` (==3); layout below is specialized for dim=3.

    float* out = (float*)d_out;
    const long long half = (long long)BDIM * CDIM * HDIM * WHALF; // s size
    float* out_s = out;
    float* out_d = out + half;

    const unsigned grid = (unsigned)(TOTAL_ROWS / ROWS_PER_BLOCK); // 65536
    lifting_dwt1d_kernel<<<grid, BLOCK, 0, stream>>>(x, Pw, Pb, Uw, Ub, out_s, out_d);
}